// RawCovTSMNet_39161511805483
// MI455X (gfx1250) — compile-verified
//
#include <hip/hip_runtime.h>
#include <hip/hip_bf16.h>
#include <math.h>
#include <stdint.h>

#define C_IN 62
#define T_LEN 2000
#define BATCH 256
#define NPAD 64
#define VEC_DIM 1953
#define HIDDEN 64
#define NCLS 3
#define SPD_EPS 1e-3f
#define LOG_EPS 1e-6f
#define NSWEEP 12

#define TK 80          // timesteps per LDS chunk (2000 = 25 * 80)
#define NCHUNK 25
#define LSTR 84        // padded LDS row stride (floats): bank-conflict-free
#define F4PR 20        // float4 loads per row per chunk (80/4)

typedef __attribute__((ext_vector_type(2))) float v2f;
typedef __attribute__((ext_vector_type(8))) float v8f;

// ---------------------------------------------------------------------------
// Kernel 1: Gram matrix G[b] = X X^T (padded 64x64) + channel sums.
// One block (512 thr, 16 waves) per batch. X is staged global->LDS with
// async B128 copies (ASYNCcnt), double-buffered against the WMMA compute.
// Each wave owns one 16x16 tile; K-loop via V_WMMA_F32_16X16X4_F32.
// Pad rows 62/63 are clamped (loaded from row 61): their Gram entries are
// dead (never read downstream), so no masking is needed anywhere.
// ---------------------------------------------------------------------------
__device__ __forceinline__ void stage_chunk(const float* __restrict__ xb,
                                            float* __restrict__ buf,
                                            int t0, int tid) {
  for (int idx = tid; idx < NPAD * F4PR; idx += 512) {
    int row = idx / F4PR;
    int c4  = idx - row * F4PR;
    int rc  = row < C_IN ? row : (C_IN - 1);              // clamp: dead rows
    const float* gp = xb + (size_t)rc * T_LEN + t0 + c4 * 4;
    uint32_t la = (uint32_t)(uintptr_t)(&buf[row * LSTR + c4 * 4]);
    asm volatile("global_load_async_to_lds_b128 %0, %1, off"
                 :: "v"(la), "v"(gp) : "memory");
  }
}

__global__ __launch_bounds__(512)
void gram_kernel(const float* __restrict__ x,
                 float* __restrict__ gram,
                 float* __restrict__ sums) {
  __shared__ float Xs[2][NPAD * LSTR];
  __shared__ float red[512];

  int b   = blockIdx.x;
  int tid = threadIdx.x;
  int wave = tid >> 5;          // 0..15
  int lane = tid & 31;
  int I = wave >> 2;            // row tile
  int J = wave & 3;             // col tile
  int m = lane & 15;
  int koff = (lane >> 4) << 1;  // 0 or 2

  int rowA = I * 16 + m;        // A fragment source row in LDS
  int rowB = J * 16 + m;        // B fragment source row in LDS

  // per-thread slice for channel-sum accumulation: 8 threads per row
  int srow = tid >> 3;          // 0..63
  int ssub = (tid & 7) * 10;    // 10 elements each (80 per chunk)

  const float* xb = x + (size_t)b * C_IN * T_LEN;

  v8f acc = {0.f, 0.f, 0.f, 0.f, 0.f, 0.f, 0.f, 0.f};
  float rsum = 0.f;

  stage_chunk(xb, Xs[0], 0, tid);
  asm volatile("s_wait_asynccnt 0" ::: "memory");
  __syncthreads();

  for (int c = 0; c < NCHUNK; ++c) {
    int cb = c & 1;
    if (c + 1 < NCHUNK) stage_chunk(xb, Xs[(c + 1) & 1], (c + 1) * TK, tid);

    const float* Xc = Xs[cb];
    const float* pa = Xc + rowA * LSTR + koff;
    const float* pb = Xc + rowB * LSTR + koff;
#pragma unroll
    for (int tt = 0; tt < TK; tt += 4) {
      v2f a  = *(const v2f*)(pa + tt);
      v2f bb = *(const v2f*)(pb + tt);
      acc = __builtin_amdgcn_wmma_f32_16x16x4_f32(
          false, a, false, bb, (short)0, acc, false, false);
    }
    // fold channel sums from the staged chunk (no second HBM pass)
    const float* sp = Xc + srow * LSTR + ssub;
#pragma unroll
    for (int k = 0; k < 10; ++k) rsum += sp[k];

    asm volatile("s_wait_asynccnt 0" ::: "memory");
    __syncthreads();
  }

  // write Gram tile (C layout: lanes 0-15 -> N=lane, M=r; 16-31 -> M=r+8)
  int n  = lane & 15;
  int mb = (lane >> 4) * 8;
  float* g = gram + (size_t)b * (NPAD * NPAD);
#pragma unroll
  for (int r = 0; r < 8; ++r) {
    g[(I * 16 + mb + r) * NPAD + J * 16 + n] = acc[r];
  }

  // reduce 8 partials per channel row
  red[tid] = rsum;
  __syncthreads();
  if ((tid & 7) == 0) {
    float s = 0.f;
#pragma unroll
    for (int k = 0; k < 8; ++k) s += red[tid + k];
    sums[b * NPAD + srow] = s;
  }
}

// ---------------------------------------------------------------------------
// Kernel 2: cov[b] = W * ((G - s s^T/T)/(T-1)) * W^T + eps*I, padded with
// identity in the 62..63 block. One block per batch.
// ---------------------------------------------------------------------------
__global__ __launch_bounds__(256)
void covtrans_kernel(const float* __restrict__ gram,
                     const float* __restrict__ sums,
                     const float* __restrict__ conv_w,
                     float* __restrict__ covout) {
  int b = blockIdx.x;
  int tid = threadIdx.x;
  __shared__ float Sm[NPAD * NPAD];
  __shared__ float M1[NPAD * NPAD];
  __shared__ float Wm[C_IN * NPAD];
  __shared__ float sloc[NPAD];

  if (tid < NPAD) sloc[tid] = sums[b * NPAD + tid];
  __syncthreads();

  const float invT  = 1.0f / (float)T_LEN;
  const float invT1 = 1.0f / (float)(T_LEN - 1);
  const float* g = gram + (size_t)b * (NPAD * NPAD);

  for (int idx = tid; idx < NPAD * NPAD; idx += 256) {
    int i = idx >> 6, j = idx & 63;
    float v = 0.f;
    if (i < C_IN && j < C_IN)
      v = (g[idx] - sloc[i] * sloc[j] * invT) * invT1;
    Sm[idx] = v;
  }
  for (int idx = tid; idx < C_IN * NPAD; idx += 256) {
    int i = idx >> 6, j = idx & 63;
    Wm[idx] = (j < C_IN) ? conv_w[i * C_IN + j] : 0.f;
  }
  __syncthreads();

  // M1 = W * S
  for (int idx = tid; idx < C_IN * NPAD; idx += 256) {
    int i = idx >> 6, j = idx & 63;
    float acc = 0.f;
    if (j < C_IN) {
      for (int k = 0; k < C_IN; ++k) acc += Wm[i * NPAD + k] * Sm[k * NPAD + j];
    }
    M1[idx] = acc;
  }
  __syncthreads();

  // cov = M1 * W^T (+ eps I), pad block = identity
  float* co = covout + (size_t)b * (NPAD * NPAD);
  for (int idx = tid; idx < NPAD * NPAD; idx += 256) {
    int i = idx >> 6, j = idx & 63;
    float v;
    if (i < C_IN && j < C_IN) {
      float acc = 0.f;
      for (int k = 0; k < C_IN; ++k) acc += M1[i * NPAD + k] * Wm[j * NPAD + k];
      if (i == j) acc += SPD_EPS;
      v = acc;
    } else {
      v = (i == j) ? 1.0f : 0.0f;
    }
    co[idx] = v;
  }
}

// ---------------------------------------------------------------------------
// Kernel 3: parallel Jacobi eigensolver (Brent-Luk round-robin) + log-map
// reconstruction + upper-tri vectorization. One block (256 thr) per batch.
// ---------------------------------------------------------------------------
__device__ __forceinline__ void get_pair(int r, int k, int& p, int& q) {
  int a, c;
  if (k == 0) { a = 63; c = r; }
  else {
    a = (r + k) % 63;
    c = (r - k + 63) % 63;
  }
  p = a < c ? a : c;
  q = a < c ? c : a;
}

__global__ __launch_bounds__(256)
void jacobi_kernel(const float* __restrict__ covin, float* __restrict__ vecout) {
  int b = blockIdx.x;
  int tid = threadIdx.x;
  __shared__ float A[NPAD * NPAD];
  __shared__ float Vm[NPAD * NPAD];
  __shared__ float cs[64];   // c in [0..31], s in [32..63]
  __shared__ float ev[NPAD];

  const float* co = covin + (size_t)b * (NPAD * NPAD);
  for (int idx = tid; idx < NPAD * NPAD; idx += 256) {
    int i = idx >> 6, j = idx & 63;
    A[idx]  = 0.5f * (co[i * NPAD + j] + co[j * NPAD + i]);  // symmetrize
    Vm[idx] = (i == j) ? 1.0f : 0.0f;
  }
  __syncthreads();

  for (int sweep = 0; sweep < NSWEEP; ++sweep) {
    for (int r = 0; r < 63; ++r) {
      if (tid < 32) {
        int p, q; get_pair(r, tid, p, q);
        float apq = A[p * NPAD + q];
        float c = 1.f, s = 0.f;
        if (fabsf(apq) > 1e-12f) {
          float tau = (A[q * NPAD + q] - A[p * NPAD + p]) / (2.f * apq);
          float t = (tau >= 0.f ? 1.f : -1.f) /
                    (fabsf(tau) + sqrtf(1.f + tau * tau));
          c = rsqrtf(1.f + t * t);
          s = t * c;
        }
        cs[tid] = c; cs[32 + tid] = s;
      }
      __syncthreads();
      // row update: A <- J^T A
      for (int it = 0; it < 8; ++it) {
        int idx = tid + it * 256;     // 32 pairs x 64 cols
        int k = idx >> 6, j = idx & 63;
        int p, q; get_pair(r, k, p, q);
        float c = cs[k], s = cs[32 + k];
        float ap = A[p * NPAD + j], aq = A[q * NPAD + j];
        A[p * NPAD + j] = c * ap - s * aq;
        A[q * NPAD + j] = s * ap + c * aq;
      }
      __syncthreads();
      // column update: A <- A J ; V <- V J
      for (int it = 0; it < 8; ++it) {
        int idx = tid + it * 256;     // 32 pairs x 64 rows
        int k = idx >> 6, i = idx & 63;
        int p, q; get_pair(r, k, p, q);
        float c = cs[k], s = cs[32 + k];
        float ap = A[i * NPAD + p], aq = A[i * NPAD + q];
        A[i * NPAD + p] = c * ap - s * aq;
        A[i * NPAD + q] = s * ap + c * aq;
        float vp = Vm[i * NPAD + p], vq = Vm[i * NPAD + q];
        Vm[i * NPAD + p] = c * vp - s * vq;
        Vm[i * NPAD + q] = s * vp + c * vq;
      }
      __syncthreads();
    }
  }

  if (tid < NPAD) ev[tid] = logf(fmaxf(A[tid * NPAD + tid], LOG_EPS));
  __syncthreads();
  for (int idx = tid; idx < NPAD * NPAD; idx += 256)
    A[idx] = Vm[idx] * ev[idx & 63];
  __syncthreads();

  // upper-tri (row-major, incl. diag) of  Vl * V^T  for i,j < 62
  float* vout = vecout + (size_t)b * VEC_DIM;
  for (int e = tid; e < VEC_DIM; e += 256) {
    int i = (int)((125.0f - sqrtf(15625.0f - 8.0f * (float)e)) * 0.5f);
    if (i < 0) i = 0;
    while (i > 0 && i * (125 - i) / 2 > e) --i;
    while ((i + 1) * (125 - (i + 1)) / 2 <= e) ++i;
    int j = i + (e - i * (125 - i) / 2);
    float acc = 0.f;
    for (int k = 0; k < NPAD; ++k)
      acc += A[i * NPAD + k] * Vm[j * NPAD + k];
    vout[e] = acc;
  }
}

// ---------------------------------------------------------------------------
// Kernel 4: feat = relu(vec @ proj_w^T + proj_b); logits = feat @ head_w^T + b
// ---------------------------------------------------------------------------
__global__ __launch_bounds__(64)
void head_kernel(const float* __restrict__ vec,
                 const float* __restrict__ proj_w,
                 const float* __restrict__ proj_b,
                 const float* __restrict__ head_w,
                 const float* __restrict__ head_b,
                 float* __restrict__ out) {
  int b = blockIdx.x;
  int h = threadIdx.x;   // 0..63
  __shared__ float feat[HIDDEN];
  const float* v = vec + (size_t)b * VEC_DIM;
  const float* wrow = proj_w + (size_t)h * VEC_DIM;
  float acc = proj_b[h];
  for (int e = 0; e < VEC_DIM; ++e) acc += v[e] * wrow[e];
  feat[h] = fmaxf(acc, 0.f);
  __syncthreads();
  if (h < NCLS) {
    float l = head_b[h];
    for (int k = 0; k < HIDDEN; ++k) l += feat[k] * head_w[h * HIDDEN + k];
    out[b * NCLS + h] = l;
  }
}

// ---------------------------------------------------------------------------
extern "C" void kernel_launch(void* const* d_in, const int* in_sizes, int n_in,
                              void* d_out, int out_size, void* d_ws, size_t ws_size,
                              hipStream_t stream) {
  const float* x      = (const float*)d_in[0];
  const float* conv_w = (const float*)d_in[1];
  // d_in[2] = conv_b: cancels under centering (cov(Wx+b) = W cov(x) W^T)
  const float* proj_w = (const float*)d_in[3];
  const float* proj_b = (const float*)d_in[4];
  const float* head_w = (const float*)d_in[5];
  const float* head_b = (const float*)d_in[6];
  float* out = (float*)d_out;

  float* ws   = (float*)d_ws;
  float* sums = ws;                                   // 256*64
  float* gram = sums + BATCH * NPAD;                  // 256*4096
  float* cov  = gram + (size_t)BATCH * NPAD * NPAD;   // 256*4096
  float* vecb = cov  + (size_t)BATCH * NPAD * NPAD;   // 256*1953

  gram_kernel    <<<BATCH, 512, 0, stream>>>(x, gram, sums);
  covtrans_kernel<<<BATCH, 256, 0, stream>>>(gram, sums, conv_w, cov);
  jacobi_kernel  <<<BATCH, 256, 0, stream>>>(cov, vecb);
  head_kernel    <<<BATCH,  64, 0, stream>>>(vecb, proj_w, proj_b, head_w, head_b, out);
}